// CapsNet_395136991692
// MI455X (gfx1250) — compile-verified
//
#include <hip/hip_runtime.h>
#include <hip/hip_bf16.h>
#include <math.h>

// Shapes from the reference
#define BB   64
#define SS   256
#define DD   768
#define CAPN 8
#define NT   10
#define KK   2048   // S*CAP

typedef float v2f __attribute__((ext_vector_type(2)));
typedef float v8f __attribute__((ext_vector_type(8)));

// ---------------------------------------------------------------------------
// Kernel 1: sem = x @ fc1^T (+bias), squash over NT, store u[r][b][k]
// GEMM: M=B*S=16384, N=80 (col j = c*NT + n), K=768, fp32 WMMA 16x16x4.
// Block = 5 waves; wave w computes the 16x16 tile for cols [16w, 16w+16).
// ---------------------------------------------------------------------------
__global__ __launch_bounds__(160) void sem_squash_kernel(
    const float* __restrict__ x, const float* __restrict__ fc1_w,
    const float* __restrict__ fc1_b, float* __restrict__ u) {
  __shared__ float sem[16][80];
  const int mtile = blockIdx.x;            // 0..1023 (M/16)
  const int tid   = threadIdx.x;
  const int w     = tid >> 5;              // ntile 0..4
  const int lane  = tid & 31;
  const int lhalf = lane & 15;
  const int koff  = (lane < 16) ? 0 : 2;
  const int col   = w * 16 + lhalf;        // 0..79 ; j = c*NT + n
  const int cc    = col / NT;
  const int nn    = col - cc * NT;
  const float* xrow = x + (size_t)(mtile * 16 + lhalf) * DD;
  const float* wrow = fc1_w + (size_t)(nn * CAPN + cc) * DD;   // fc1_w[n][c][:]

  v8f acc = {};
  #pragma unroll 4
  for (int k4 = 0; k4 < DD; k4 += 4) {
    v2f a = *reinterpret_cast<const v2f*>(xrow + k4 + koff);   // A: x rows
    v2f b = *reinterpret_cast<const v2f*>(wrow + k4 + koff);   // B = W'^T
    acc = __builtin_amdgcn_wmma_f32_16x16x4_f32(
        false, a, false, b, (short)0, acc, false, false);
  }
  const float bias = fc1_b[nn * CAPN + cc];
  const int rbase = (lane < 16) ? 0 : 8;
  #pragma unroll
  for (int v = 0; v < 8; ++v) sem[rbase + v][col] = acc[v] + bias;
  __syncthreads();

  // squash over NT for each of 16 rows x 8 capsules
  if (tid < 128) {
    const int row = tid >> 3, c = tid & 7;
    float t[NT];
    float sq = 0.f;
    #pragma unroll
    for (int n = 0; n < NT; ++n) { t[n] = sem[row][c * NT + n]; sq += t[n] * t[n]; }
    const float scale = sq / ((1.f + sq) * sqrtf(sq));
    const int m = mtile * 16 + row;        // tiles never straddle a batch row
    const int b = m >> 8, s = m & (SS - 1);
    const int k = s * CAPN + c;
    #pragma unroll
    for (int n = 0; n < NT; ++n)
      u[((size_t)n * BB + b) * KK + k] = t[n] * scale;   // u[r][b][k]
  }
}

// ---------------------------------------------------------------------------
// Kernel 2: priors[c][r][b][l] = u[r] (64x2048) @ rw[c][r] (2048x256)
// fp32 WMMA 16x16x4; grid (8 n-blocks, NT, CAP); r > task exits (exp(-10000)
// underflows to exactly 0 in the softmax, so those routes never matter).
// ---------------------------------------------------------------------------
__global__ __launch_bounds__(256) void priors_kernel(
    const float* __restrict__ u, const float* __restrict__ rw,
    const int* __restrict__ taskp, float* __restrict__ priors) {
  const int r = blockIdx.y, c = blockIdx.z;
  if (r > taskp[0]) return;
  const int tid = threadIdx.x;
  const int w = tid >> 5, lane = tid & 31, lhalf = lane & 15;
  const int koff  = (lane < 16) ? 0 : 2;
  const int mtile = w & 3;                       // 4 M-tiles (B=64)
  const int ntile = blockIdx.x * 2 + (w >> 2);   // 16 N-tiles (S=256)
  const int col   = ntile * 16 + lhalf;

  const float* arow = u + ((size_t)r * BB + mtile * 16 + lhalf) * KK;
  const float* bp   = rw + (size_t)(c * NT + r) * KK * SS;

  v8f acc = {};
  #pragma unroll 4
  for (int k4 = 0; k4 < KK; k4 += 4) {
    v2f a = *reinterpret_cast<const v2f*>(arow + k4 + koff);
    v2f b;
    b.x = bp[(size_t)(k4 + koff) * SS + col];
    b.y = bp[(size_t)(k4 + koff + 1) * SS + col];
    acc = __builtin_amdgcn_wmma_f32_16x16x4_f32(
        false, a, false, b, (short)0, acc, false, false);
  }
  float* dst = priors + (size_t)(c * NT + r) * BB * SS;
  const int rbase = mtile * 16 + ((lane < 16) ? 0 : 8);
  #pragma unroll
  for (int v = 0; v < 8; ++v)
    dst[(size_t)(rbase + v) * SS + col] = acc[v];
}

// ---------------------------------------------------------------------------
// Kernel 3: dynamic routing. Logit updates are scalars per (c,b,r) (sum over
// l), so routing = scalar softmax + block reductions. Output = UNsquashed vote.
// One block (256 threads, lane l) per (c,b).
// ---------------------------------------------------------------------------
__device__ __forceinline__ float wave_sum(float v) {
  #pragma unroll
  for (int off = 16; off > 0; off >>= 1) v += __shfl_xor(v, off, 32);
  return v;
}
__device__ __forceinline__ float block_sum(float v, float* red) {
  v = wave_sum(v);
  const int wid = threadIdx.x >> 5;
  if ((threadIdx.x & 31) == 0) red[wid] = v;
  __syncthreads();
  float s = 0.f;
  #pragma unroll
  for (int i = 0; i < 8; ++i) s += red[i];
  __syncthreads();
  return s;
}

__global__ __launch_bounds__(256) void routing_kernel(
    const float* __restrict__ priors, const int* __restrict__ taskp,
    float* __restrict__ vote) {
  __shared__ float red[8];
  const int b = blockIdx.x, c = blockIdx.y;
  const int l = threadIdx.x;
  const int task = taskp[0];

  float p[NT];
  #pragma unroll
  for (int r = 0; r < NT; ++r)
    p[r] = (r <= task) ? priors[((size_t)(c * NT + r) * BB + b) * SS + l] : 0.f;

  float a[NT];
  #pragma unroll
  for (int r = 0; r < NT; ++r) a[r] = 0.f;

  float vote_l = 0.f;
  for (int it = 0; it < 3; ++it) {
    // softmax over active routes (uniform scalars across the block)
    float mx = -3.4e38f;
    for (int r = 0; r < NT; ++r) if (r <= task && a[r] > mx) mx = a[r];
    float e[NT], denom = 0.f;
    for (int r = 0; r < NT; ++r) {
      e[r] = (r <= task) ? expf(a[r] - mx) : 0.f;
      denom += e[r];
    }
    vote_l = 0.f;
    for (int r = 0; r < NT; ++r) vote_l += (e[r] / denom) * p[r];

    if (it < 2) {
      const float sq = block_sum(vote_l * vote_l, red);
      const float scale = sq / ((1.f + sq) * sqrtf(sq));
      const float out_l = vote_l * scale;
      for (int r = 0; r < NT; ++r)          // unconditional: uniform barriers
        a[r] += block_sum(p[r] * out_l, red);
    }
  }
  vote[((size_t)c * BB + b) * SS + l] = vote_l;   // flat [c][b][l]
}

// ---------------------------------------------------------------------------
// Kernel 4: out[m][d] = sum_c h[m][c]*larger_w[d][c] + larger_b[d]
// h = flat reinterpret of vote buffer as [16384][8]. One block per m.
// ---------------------------------------------------------------------------
__global__ __launch_bounds__(256) void out_kernel(
    const float* __restrict__ vote, const float* __restrict__ lw,
    const float* __restrict__ lb, float* __restrict__ out) {
  __shared__ float h[CAPN];
  const int m = blockIdx.x;
  if (threadIdx.x < CAPN) h[threadIdx.x] = vote[(size_t)m * CAPN + threadIdx.x];
  __syncthreads();
  for (int d = threadIdx.x; d < DD; d += 256) {
    const float* wr = lw + (size_t)d * CAPN;
    float acc = lb[d];
    #pragma unroll
    for (int c = 0; c < CAPN; ++c) acc += h[c] * wr[c];
    out[(size_t)m * DD + d] = acc;
  }
}

// ---------------------------------------------------------------------------
extern "C" void kernel_launch(void* const* d_in, const int* in_sizes, int n_in,
                              void* d_out, int out_size, void* d_ws, size_t ws_size,
                              hipStream_t stream) {
  const float* x        = (const float*)d_in[0];
  const int*   task     = (const int*)d_in[1];
  const float* fc1_w    = (const float*)d_in[2];
  const float* fc1_b    = (const float*)d_in[3];
  const float* route_w  = (const float*)d_in[4];
  const float* larger_w = (const float*)d_in[5];
  const float* larger_b = (const float*)d_in[6];
  float* out = (float*)d_out;

  float* ws        = (float*)d_ws;
  float* u_ws      = ws;                         // NT*B*KK   = 1,310,720 f
  float* priors_ws = ws + (size_t)NT * BB * KK;  // CAP*NT*B*S= 1,310,720 f
  float* vote_ws   = priors_ws + (size_t)CAPN * NT * BB * SS;   // 131,072 f

  sem_squash_kernel<<<(BB * SS) / 16, 160, 0, stream>>>(x, fc1_w, fc1_b, u_ws);
  priors_kernel<<<dim3(8, NT, CAPN), 256, 0, stream>>>(u_ws, route_w, task, priors_ws);
  routing_kernel<<<dim3(BB, CAPN), 256, 0, stream>>>(priors_ws, task, vote_ws);
  out_kernel<<<BB * SS, 256, 0, stream>>>(vote_ws, larger_w, larger_b, out);
}